// Propagate_Planes_Layer_MatrixBroadband_21698174779563
// MI455X (gfx1250) — compile-verified
//
#include <hip/hip_runtime.h>
#include <math.h>

// ---------------------------------------------------------------------------
// Angular-spectrum propagation for MI455X (gfx1250, wave32).
// 1024-pt FFT = radix-16 (WMMA) x radix-16 (WMMA) x radix-4 (VALU), in LDS.
// Each radix-16 stage applies the dense 16x16 DFT matrix via
// V_WMMA_F32_16X16X4_F32 (4 K-chunks x 4 real matmuls = 16 WMMA per batch).
// All twiddles come from a per-block 1024-entry LDS cos/sin table
// (theta_p = -2*pi*p/1024); the inverse transform conjugates the table.
// ---------------------------------------------------------------------------

#define L_DIM 16
#define B_DIM 2
#define N_DIM 512
#define NP    1024
#define PAD   256
#define DXF   5e-07f
#define ZF    0.001f

typedef __attribute__((ext_vector_type(2))) float v2f;
typedef __attribute__((ext_vector_type(8))) float v8f;

// Only check the builtin in the *device* pass: AMDGCN builtins are not
// declared when clang targets the host.
#if defined(__HIP_DEVICE_COMPILE__) && defined(__gfx1250__)
#if !__has_builtin(__builtin_amdgcn_wmma_f32_16x16x4_f32)
#error "wmma_f32_16x16x4_f32 builtin not available on this toolchain"
#endif
#endif

__device__ __forceinline__ v8f wmma4(v2f a, v2f b, v8f c) {
  // D(16x16,f32) = A(16x4,f32) * B(4x16,f32) + C
  return __builtin_amdgcn_wmma_f32_16x16x4_f32(
      /*neg_a=*/false, a, /*neg_b=*/false, b,
      /*c_mod=*/(short)0, c, /*reuse_a=*/false, /*reuse_b=*/false);
}

#define PI_F 3.14159265358979323846f

// fftfreq(1024, d=DX) in cycles/m
__device__ __forceinline__ float fftfreq1024(int k) {
  float kk = (k < 512) ? (float)k : (float)(k - 1024);
  return kk / (1024.0f * DXF);
}

// Build the forward twiddle table: twc[p] = cos(-2pi p/1024),
// tws[p] = sin(-2pi p/1024). One sincosf per entry, once per block.
__device__ __forceinline__ void init_twiddles(float* __restrict__ twc,
                                              float* __restrict__ tws,
                                              int tid, int nthreads) {
  for (int p = tid; p < NP; p += nthreads) {
    float sn, cs;
    sincosf(-2.0f * PI_F * (float)p / 1024.0f, &sn, &cs);
    twc[p] = cs;
    tws[p] = sn;
  }
}

// ---------------------------------------------------------------------------
// In-place 1024-point complex FFT over sre/sim (1024 floats each in LDS).
// Executed cooperatively by a 128-thread group (4 waves). `tid` in [0,128).
// inv = 0 forward, 1 inverse (unnormalized).
//
// Four-step: n = 64*r + s, X[k1 + 16*k'] with
//   stage A : 64 x DFT16 over r, twiddle w_1024^{s*k1}
//   stage B1: per 64-block, 4 x DFT16 over r', twiddle w_64^{s'*k1'}
//   stage B2: 256 x DFT4 over s'  -> natural-order output.
//
// WMMA operand layout assumptions (32-bit):
//   A (16x4): M = lane&15, K = vgpr + 2*(lane>>4)   [per ISA table]
//   B (4x16): N = lane&15, K = vgpr + 2*(lane>>4)   [assumed, mirrors C/D]
//   C/D (16x16): M = vgpr + 8*(lane>>4), N = lane&15 [per ISA table]
// ---------------------------------------------------------------------------
__device__ __forceinline__ void fft1024(float* __restrict__ sre,
                                        float* __restrict__ sim,
                                        const float* __restrict__ twc,
                                        const float* __restrict__ tws,
                                        int tid, int inv) {
  const float sm = inv ? -1.0f : 1.0f;  // sine multiplier (conjugate if inv)
  const int lane = tid & 31;
  const int wv   = tid >> 5;     // wave id within the group: 0..3
  const int m    = lane & 15;    // matrix row / column index
  const int hi   = lane >> 4;    // upper half of the wave

  __syncthreads();  // prior producers of sre/sim (and the twiddle table)

  // DFT-16 matrix in A-operand layout, split into 4 K-chunks, from the table:
  // W[m][k] = e^{sign*2pi*i*m*k/16} -> table index ((m*k) mod 16) * 64.
  // wr = Re(W), wi = Im(W), wni = -Im(W)  (f32 WMMA has no A/B negate).
  v2f wr[4], wi[4], wni[4];
#pragma unroll
  for (int c = 0; c < 4; ++c) {
#pragma unroll
    for (int v = 0; v < 2; ++v) {
      int k = 4 * c + v + 2 * hi;
      int p = ((m * k) & 15) << 6;
      float cs = twc[p];
      float sn = sm * tws[p];
      wr[c][v] = cs; wi[c][v] = sn; wni[c][v] = -sn;
    }
  }

  // ---------------- Stage A: DFT16 over r of x[64r+s], wave owns 16 columns
  {
    const int col = 16 * wv + m;  // s
    v2f br[4], bi[4];
#pragma unroll
    for (int c = 0; c < 4; ++c) {
#pragma unroll
      for (int v = 0; v < 2; ++v) {
        int r = 4 * c + v + 2 * hi;           // K index
        int idx = 64 * r + col;
        br[c][v] = sre[idx];
        bi[c][v] = sim[idx];
      }
    }
    v8f yr = {}, yi = {};
#pragma unroll
    for (int c = 0; c < 4; ++c) {             // interleave the two acc chains
      yr = wmma4(wr[c], br[c], yr);
      yi = wmma4(wi[c], br[c], yi);
    }
#pragma unroll
    for (int c = 0; c < 4; ++c) {
      yr = wmma4(wni[c], bi[c], yr);
      yi = wmma4(wr[c],  bi[c], yi);
    }
    // twiddle w_1024^{s*k1}, store in place (wave-private columns)
#pragma unroll
    for (int v8 = 0; v8 < 8; ++v8) {
      int k1 = v8 + 8 * hi;
      int p = (col * k1) & 1023;
      float cs = twc[p];
      float sn = sm * tws[p];
      float zr = yr[v8], zi = yi[v8];
      int idx = 64 * k1 + col;
      sre[idx] = zr * cs - zi * sn;
      sim[idx] = zr * sn + zi * cs;
    }
  }
  __syncthreads();

  // ---------------- Stage B1: per 64-block DFT16 over r' of z[4r'+s']
  {
    const int k1 = 4 * wv + (m >> 2);  // wave owns 4 contiguous 64-blocks
    const int sp = m & 3;
    v2f br[4], bi[4];
#pragma unroll
    for (int c = 0; c < 4; ++c) {
#pragma unroll
      for (int v = 0; v < 2; ++v) {
        int r = 4 * c + v + 2 * hi;           // K index = r'
        int idx = 64 * k1 + 4 * r + sp;
        br[c][v] = sre[idx];
        bi[c][v] = sim[idx];
      }
    }
    v8f yr = {}, yi = {};
#pragma unroll
    for (int c = 0; c < 4; ++c) {
      yr = wmma4(wr[c], br[c], yr);
      yi = wmma4(wi[c], br[c], yi);
    }
#pragma unroll
    for (int c = 0; c < 4; ++c) {
      yr = wmma4(wni[c], bi[c], yr);
      yi = wmma4(wr[c],  bi[c], yi);
    }
    // twiddle w_64^{s'*k1'} -> table index ((s'*k1') mod 64) * 16
#pragma unroll
    for (int v8 = 0; v8 < 8; ++v8) {
      int k1p = v8 + 8 * hi;
      int p = ((sp * k1p) & 63) << 4;
      float cs = twc[p];
      float sn = sm * tws[p];
      float zr = yr[v8], zi = yi[v8];
      int idx = 64 * k1 + 4 * k1p + sp;
      sre[idx] = zr * cs - zi * sn;
      sim[idx] = zr * sn + zi * cs;
    }
  }
  __syncthreads();

  // ---------------- Stage B2: 256 x DFT4 over s', permuting output indices
  {
    const float bsign = inv ? 1.0f : -1.0f;  // omega_4 = i * bsign
    float ur[2][4], ui[2][4];
#pragma unroll
    for (int g = 0; g < 2; ++g) {
      int gg = 2 * tid + g;                 // 0..255
      int k1 = gg >> 4, k1p = gg & 15;
#pragma unroll
      for (int sp = 0; sp < 4; ++sp) {
        int idx = 64 * k1 + 4 * k1p + sp;
        ur[g][sp] = sre[idx];
        ui[g][sp] = sim[idx];
      }
    }
    __syncthreads();  // global permutation: all reads before any writes
#pragma unroll
    for (int g = 0; g < 2; ++g) {
      int gg = 2 * tid + g;
      int k1 = gg >> 4, k1p = gg & 15;
      float a0r = ur[g][0] + ur[g][2], a0i = ui[g][0] + ui[g][2];
      float a1r = ur[g][0] - ur[g][2], a1i = ui[g][0] - ui[g][2];
      float b0r = ur[g][1] + ur[g][3], b0i = ui[g][1] + ui[g][3];
      float b1r = ur[g][1] - ur[g][3], b1i = ui[g][1] - ui[g][3];
      // w = i*bsign ; w*(x+iy) = bsign*(-y + ix)
      float wbr = -bsign * b1i, wbi = bsign * b1r;
      int base = k1 + 16 * k1p;             // + 256*k2'
      sre[base +   0] = a0r + b0r; sim[base +   0] = a0i + b0i;
      sre[base + 256] = a1r + wbr; sim[base + 256] = a1i + wbi;
      sre[base + 512] = a0r - b0r; sim[base + 512] = a0i - b0i;
      sre[base + 768] = a1r - wbr; sim[base + 768] = a1i - wbi;
    }
  }
  __syncthreads();
}

// ---------------------------------------------------------------------------
// Pass 1: build padded complex field (amp * aperture * e^{i phase}) and
// forward-FFT every row. Rows outside the central band are exactly zero.
// grid (1024, 32), block 128.
// ---------------------------------------------------------------------------
__global__ void __launch_bounds__(128)
asp_rows_fwd(const float* __restrict__ amp, const float* __restrict__ ph,
             const float* __restrict__ ap, float2* __restrict__ U) {
  __shared__ float sre[NP];
  __shared__ float sim[NP];
  __shared__ float twc[NP];
  __shared__ float tws[NP];
  const int y   = blockIdx.x;
  const int img = blockIdx.y;   // l*B + b
  const int tid = threadIdx.x;
  float2* row = U + ((size_t)img * NP + y) * NP;

  if (y < PAD || y >= PAD + N_DIM) {
#pragma unroll
    for (int j = 0; j < 8; ++j) row[tid + 128 * j] = make_float2(0.f, 0.f);
    return;
  }
  init_twiddles(twc, tws, tid, 128);

  const int ys = y - PAD;
  const float* arow  = amp + ((size_t)img * N_DIM + ys) * N_DIM;
  const float* prow  = ph  + ((size_t)img * N_DIM + ys) * N_DIM;
  const float* aprow = ap  + (size_t)ys * N_DIM;

#pragma unroll
  for (int j = 0; j < 8; ++j) {
    int x = tid + 128 * j;
    float re = 0.f, im = 0.f;
    if (x >= PAD && x < PAD + N_DIM) {
      int xs = x - PAD;
      float a = arow[xs] * aprow[xs];
      float sn, cs;
      sincosf(prow[xs], &sn, &cs);
      re = a * cs;
      im = a * sn;
    }
    sre[x] = re;
    sim[x] = im;
  }
  fft1024(sre, sim, twc, tws, tid, 0);
#pragma unroll
  for (int j = 0; j < 8; ++j) {
    int x = tid + 128 * j;
    row[x] = make_float2(sre[x], sim[x]);
  }
}

// ---------------------------------------------------------------------------
// Pass 2: forward column FFT, multiply by H(lambda,fx,fy) (ifft 1/2^20
// normalization folded in), inverse column FFT. 4 columns per block.
// grid (256, 32), block 512, 40 KB static LDS.
// ---------------------------------------------------------------------------
__global__ void __launch_bounds__(512)
asp_cols_H(float2* __restrict__ U, const float* __restrict__ wavelengths) {
  __shared__ float smem[4 * 2 * NP];  // [col][re|im][1024]
  __shared__ float twc[NP];
  __shared__ float tws[NP];
  const int tid = threadIdx.x;        // 0..511
  const int grp = tid >> 7;           // column group 0..3
  const int tl  = tid & 127;
  const int x0  = blockIdx.x * 4;
  const int img = blockIdx.y;
  const int l   = img >> 1;           // img = l*B + b, B = 2
  float* sre = smem + grp * 2048;
  float* sim = sre + NP;
  float2* base = U + (size_t)img * NP * NP + x0;

  init_twiddles(twc, tws, tid, 512);

  // cooperative, x-contiguous load of the 4-column tile
#pragma unroll
  for (int j = 0; j < 8; ++j) {
    int e = tid + 512 * j;            // 0..4095
    int yy = e >> 2, c = e & 3;
    float2 v = base[(size_t)yy * NP + c];
    smem[c * 2048 + yy]      = v.x;
    smem[c * 2048 + NP + yy] = v.y;
  }

  fft1024(sre, sim, twc, tws, tl, 0);

  const float lam = wavelengths[l];
  const float fx  = fftfreq1024(x0 + grp);
  const float fx2 = fx * fx;
  const float scale = 1.0f / (1024.0f * 1024.0f);  // ifft2 normalization
#pragma unroll
  for (int j = 0; j < 8; ++j) {
    int ky = tl + 128 * j;
    float fy  = fftfreq1024(ky);
    float arg = 1.0f - lam * lam * (fx2 + fy * fy);
    float re = sre[ky], im = sim[ky];
    if (arg > 0.0f) {
      float kz = (2.0f * PI_F / lam) * sqrtf(fmaxf(arg, 1e-12f));
      float sn, cs;
      sincosf(ZF * kz, &sn, &cs);
      float hr = cs * scale, hi = sn * scale;
      sre[ky] = re * hr - im * hi;
      sim[ky] = re * hi + im * hr;
    } else {
      sre[ky] = 0.f;
      sim[ky] = 0.f;
    }
  }

  fft1024(sre, sim, twc, tws, tl, 1);  // leading barrier covers H writes

#pragma unroll
  for (int j = 0; j < 8; ++j) {
    int e = tid + 512 * j;
    int yy = e >> 2, c = e & 3;
    base[(size_t)yy * NP + c] =
        make_float2(smem[c * 2048 + yy], smem[c * 2048 + NP + yy]);
  }
}

// ---------------------------------------------------------------------------
// Pass 3: inverse row FFT for the 512 output rows, crop, abs/angle.
// grid (512, 32), block 128.
// ---------------------------------------------------------------------------
__global__ void __launch_bounds__(128)
asp_rows_inv(const float2* __restrict__ U, float* __restrict__ out_abs,
             float* __restrict__ out_ang) {
  __shared__ float sre[NP];
  __shared__ float sim[NP];
  __shared__ float twc[NP];
  __shared__ float tws[NP];
  const int yo  = blockIdx.x;   // 0..511
  const int img = blockIdx.y;
  const int tid = threadIdx.x;
  const float2* row = U + ((size_t)img * NP + (yo + PAD)) * NP;

  init_twiddles(twc, tws, tid, 128);

#pragma unroll
  for (int j = 0; j < 8; ++j) {
    int x = tid + 128 * j;
    float2 v = row[x];
    sre[x] = v.x;
    sim[x] = v.y;
  }
  fft1024(sre, sim, twc, tws, tid, 1);

  float* oa = out_abs + ((size_t)img * N_DIM + yo) * N_DIM;
  float* op = out_ang + ((size_t)img * N_DIM + yo) * N_DIM;
#pragma unroll
  for (int j = 0; j < 4; ++j) {
    int xo = tid + 128 * j;    // 0..511
    float re = sre[xo + PAD], im = sim[xo + PAD];
    oa[xo] = sqrtf(re * re + im * im);
    op[xo] = atan2f(im, re);
  }
}

// ---------------------------------------------------------------------------
extern "C" void kernel_launch(void* const* d_in, const int* in_sizes, int n_in,
                              void* d_out, int out_size, void* d_ws,
                              size_t ws_size, hipStream_t stream) {
  const float* amp = (const float*)d_in[0];  // (L,B,N,N) f32
  const float* ph  = (const float*)d_in[1];  // (L,B,N,N) f32
  const float* ap  = (const float*)d_in[2];  // (1,N,N)   f32
  const float* wl  = (const float*)d_in[3];  // (L,)      f32
  float* out = (float*)d_out;                // abs | angle, each L*B*N*N
  float2* U = (float2*)d_ws;                 // (L*B, 1024, 1024) complex64

  const size_t half = (size_t)L_DIM * B_DIM * N_DIM * N_DIM;  // 8388608

  asp_rows_fwd<<<dim3(NP, L_DIM * B_DIM), 128, 0, stream>>>(amp, ph, ap, U);
  asp_cols_H<<<dim3(NP / 4, L_DIM * B_DIM), 512, 0, stream>>>(U, wl);
  asp_rows_inv<<<dim3(N_DIM, L_DIM * B_DIM), 128, 0, stream>>>(U, out,
                                                               out + half);
}